// DistanceWeightedSoftIoULoss_63153199120772
// MI455X (gfx1250) — compile-verified
//
#include <hip/hip_runtime.h>
#include <hip/hip_bf16.h>
#include <cstdint>

#define BB 16
#define HH 512
#define WW 512
#define NPIX (BB * HH * WW)
#define ROWS (BB * HH)
#define BLK 256
#define NWAVE (BLK / 32)

typedef __attribute__((ext_vector_type(2))) float v2f;
typedef __attribute__((ext_vector_type(8))) float v8f;

// ---------------------------------------------------------------------------
// Kernel 0: zero per-image foreground flags
// ---------------------------------------------------------------------------
__global__ void dw_init_flags(int* __restrict__ hasfg) {
    if (threadIdx.x < BB) hasfg[threadIdx.x] = 0;
}

// ---------------------------------------------------------------------------
// Kernel 1: exact 1D EDT along H per (b,w) column (input is binary, so two
// linear sweeps are exact). Writes squared column distance; empty columns
// saturate to (1e6)^2 == 1e12 == reference INF (same fp32 rounding).
// ---------------------------------------------------------------------------
__global__ void dw_col_edt(const float* __restrict__ targets,
                           float* __restrict__ f1,
                           int* __restrict__ hasfg) {
    int idx = blockIdx.x * blockDim.x + threadIdx.x;  // 0 .. B*W-1
    if (idx >= BB * WW) return;
    int b = idx / WW, w = idx % WW;
    const size_t base = (size_t)b * HH * WW + (size_t)w;

    float d = 1.0e6f;
    int any = 0;
    for (int i = 0; i < HH; ++i) {
        float t = targets[base + (size_t)i * WW];
        if (t > 0.0f) { d = 0.0f; any = 1; }
        else          { d = fminf(d + 1.0f, 1.0e6f); }
        f1[base + (size_t)i * WW] = d;
    }
    d = 1.0e6f;
    for (int i = HH - 1; i >= 0; --i) {
        float t = targets[base + (size_t)i * WW];
        if (t > 0.0f) d = 0.0f;
        else          d = fminf(d + 1.0f, 1.0e6f);
        float m = fminf(f1[base + (size_t)i * WW], d);
        f1[base + (size_t)i * WW] = m * m;  // exact: integer <= 511, or 1e12
    }
    if (any) atomicOr(&hasfg[b], 1);  // idempotent -> deterministic
}

// ---------------------------------------------------------------------------
// Exact wave-level sum of two quantities using V_WMMA_F32_16X16X4_F32.
// A (16x4): lane L contributes (qa,qb) as A[L&15][L<16 ? 0,1 : 2,3].
// B (4x16): col0 = [1,0,1,0], col1 = [0,1,0,1], rest zero -> fp32-exact.
// C[r][0] = qa(r)+qa(r+16); C[r][1] = qb(r)+qb(r+16). Lane 0/16 hold col 0
// rows 0-7 / 8-15 in their 8 C VGPRs; one shfl_down(16) finishes the sum.
// Result: lane 0 -> total qa, lane 1 -> total qb. EXEC must be all ones.
// ---------------------------------------------------------------------------
__device__ __forceinline__ float dw_wave_pair_reduce(float qa, float qb, unsigned lane) {
#if defined(__gfx1250__) && __has_builtin(__builtin_amdgcn_wmma_f32_16x16x4_f32)
    v2f a; a.x = qa; a.y = qb;
    unsigned l16 = lane & 15u;
    v2f bm; bm.x = (l16 == 0u) ? 1.0f : 0.0f;
            bm.y = (l16 == 1u) ? 1.0f : 0.0f;
    v8f c = {};
    c = __builtin_amdgcn_wmma_f32_16x16x4_f32(false, a, false, bm,
                                              (short)0, c, false, false);
    float s = c[0] + c[1] + c[2] + c[3] + c[4] + c[5] + c[6] + c[7];
    s += __shfl_down(s, 16, 32);
    return s;
#else
    float sa = qa, sb = qb;
    for (int off = 16; off > 0; off >>= 1) {
        sa += __shfl_down(sa, off, 32);
        sb += __shfl_down(sb, off, 32);
    }
    sb = __shfl(sb, 0, 32);
    float s = 0.0f;
    if (lane == 0u) s = sa;
    if (lane == 1u) s = sb;
    return s;
#endif
}

// ---------------------------------------------------------------------------
// Kernel 2: per-row min-plus parabola pass (exact, same fp32 ops as the
// reference's min_j f[j] + (w-j)^2), fused with sigmoid / weight / partial
// sums. Row staged in LDS via async-to-LDS when the toolchain exposes it.
// ---------------------------------------------------------------------------
__global__ void dw_row_pass(const float* __restrict__ logits,
                            const float* __restrict__ targets,
                            const float* __restrict__ f1,
                            const int* __restrict__ hasfg,
                            float* __restrict__ partials) {
    __shared__ float srow[WW];
    __shared__ float wred[NWAVE][4];

    const int row = blockIdx.x;           // 0 .. B*H-1
    const int b = row / HH;
    const size_t rbase = (size_t)row * WW;
    const float* rowptr = f1 + rbase;

#if defined(__gfx1250__) && __has_builtin(__builtin_amdgcn_global_load_async_to_lds_b32) && __has_builtin(__builtin_amdgcn_s_wait_asynccnt)
    for (int w = threadIdx.x; w < WW; w += BLK) {
        __builtin_amdgcn_global_load_async_to_lds_b32(
            (__attribute__((address_space(1))) int*)(rowptr + w),
            (__attribute__((address_space(3))) int*)(srow + w),
            0, 0);
    }
    __builtin_amdgcn_s_wait_asynccnt(0);
    __syncthreads();
#else
    for (int w = threadIdx.x; w < WW; w += BLK) srow[w] = rowptr[w];
    __syncthreads();
#endif

    __builtin_prefetch(logits + rbase + threadIdx.x, 0, 3);
    __builtin_prefetch(targets + rbase + threadIdx.x, 0, 3);

    const int fg = hasfg[b];
    float s_pt = 0.0f, s_p = 0.0f, s_t = 0.0f, s_w = 0.0f;

    for (int w = threadIdx.x; w < WW; w += BLK) {
        float best = 3.0e38f;
#pragma unroll 8
        for (int j = 0; j < WW; ++j) {
            float dd = (float)(w - j);            // |dd| <= 511 -> dd*dd exact
            best = fminf(best, fmaf(dd, dd, srow[j]));
        }
        float dist = sqrtf(best);
        float wgt = fg ? (1.0f - expf(-dist / 50.0f)) : 0.0f;

        size_t off = rbase + (size_t)w;
        float lg = logits[off];
        float tg = targets[off];
        float pr = 1.0f / (1.0f + expf(-lg));

        s_pt = fmaf(pr, tg, s_pt);
        s_p += pr;
        s_t += tg;
        s_w += wgt * pr * (1.0f - tg);
    }

    // ---- block reduction: 2 WMMAs per wave (fp32-exact), then tiny LDS sum
    const unsigned lane = threadIdx.x & 31u;
    const unsigned wave = threadIdx.x >> 5;
    float r1 = dw_wave_pair_reduce(s_pt, s_p, lane);  // lane0: s_pt, lane1: s_p
    float r2 = dw_wave_pair_reduce(s_t,  s_w, lane);  // lane0: s_t,  lane1: s_w
    if (lane == 0u) { wred[wave][0] = r1; wred[wave][2] = r2; }
    if (lane == 1u) { wred[wave][1] = r1; wred[wave][3] = r2; }
    __syncthreads();
    if (threadIdx.x < 4) {
        float t = 0.0f;
        for (int wv = 0; wv < NWAVE; ++wv) t += wred[wv][threadIdx.x];
        partials[(size_t)blockIdx.x * 4 + threadIdx.x] = t;
    }
}

// ---------------------------------------------------------------------------
// Kernel 3: deterministic final reduction over block partials + loss compose
// ---------------------------------------------------------------------------
__global__ void dw_finalize(const float* __restrict__ partials,
                            float* __restrict__ out) {
    __shared__ float red[BLK];
    float acc[4] = {0.0f, 0.0f, 0.0f, 0.0f};
    for (int i = threadIdx.x; i < ROWS; i += BLK) {
        acc[0] += partials[(size_t)i * 4 + 0];
        acc[1] += partials[(size_t)i * 4 + 1];
        acc[2] += partials[(size_t)i * 4 + 2];
        acc[3] += partials[(size_t)i * 4 + 3];
    }
    float tot[4] = {0.0f, 0.0f, 0.0f, 0.0f};
    for (int q = 0; q < 4; ++q) {
        red[threadIdx.x] = acc[q];
        __syncthreads();
        for (int off = BLK / 2; off > 0; off >>= 1) {
            if ((int)threadIdx.x < off) red[threadIdx.x] += red[threadIdx.x + off];
            __syncthreads();
        }
        if (threadIdx.x == 0) tot[q] = red[0];
        __syncthreads();
    }
    if (threadIdx.x == 0) {
        float inter = tot[0], sp = tot[1], st = tot[2], sw = tot[3];
        float uni  = sp + st - inter;
        float soft = 1.0f - (inter + 1e-6f) / (uni + 1e-6f);
        float pen  = sw / (float)NPIX;
        out[0] = soft + 0.5f * pen;
    }
}

// ---------------------------------------------------------------------------
extern "C" void kernel_launch(void* const* d_in, const int* in_sizes, int n_in,
                              void* d_out, int out_size, void* d_ws, size_t ws_size,
                              hipStream_t stream) {
    (void)in_sizes; (void)n_in; (void)out_size; (void)ws_size;
    const float* logits  = (const float*)d_in[0];
    const float* targets = (const float*)d_in[1];
    float* out = (float*)d_out;

    char* ws = (char*)d_ws;
    const size_t f1_bytes = (size_t)NPIX * sizeof(float);   // 16 MiB
    float* f1       = (float*)ws;
    int*   hasfg    = (int*)(ws + f1_bytes);
    float* partials = (float*)(ws + f1_bytes + 256);        // ROWS*4 floats

    dw_init_flags<<<1, 32, 0, stream>>>(hasfg);
    dw_col_edt<<<(BB * WW + BLK - 1) / BLK, BLK, 0, stream>>>(targets, f1, hasfg);
    dw_row_pass<<<ROWS, BLK, 0, stream>>>(logits, targets, f1, hasfg, partials);
    dw_finalize<<<1, BLK, 0, stream>>>(partials, out);
}